// PositionalEncoding_85641647882476
// MI455X (gfx1250) — compile-verified
//
#include <hip/hip_runtime.h>
#include <math.h>

// out[b,s,d] = x[b,s,d]*sqrt(D) + (d even ? sin(pos*dt) : cos(pos*dt)),
// dt = exp((d&~1) * (-ln(1e4)/D)).  Memory-bound streaming kernel:
// 268 MB total traffic -> ~11.5us floor at 23.3 TB/s.

typedef float v4f __attribute__((ext_vector_type(4)));

#define PE_THREADS 256
#define PE_VEC 8   // floats per thread = 4 (sin,cos) pairs = two b128 loads/stores

__global__ __launch_bounds__(PE_THREADS)
void PositionalEncoding_gfx1250_kernel(const float* __restrict__ x,
                                       const int* __restrict__ positions,
                                       float* __restrict__ out,
                                       long long n, int log2d, int dmask,
                                       float negLogOverD, float sqrtD) {
  const long long i0 =
      ((long long)blockIdx.x * PE_THREADS + threadIdx.x) * (long long)PE_VEC;
  if (i0 >= n) return;

  // D is a power of two: shift/mask instead of the div/mod clique.
  const int row = (int)(i0 >> log2d);        // flat (b*S + s) index
  const int d0  = (int)i0 & dmask;           // even, multiple of 8

  const float posf = (float)positions[row];

  // One-shot streaming data: non-temporal 128-bit loads (don't pollute L2).
  const v4f x0 = __builtin_nontemporal_load((const v4f*)(x + i0));
  const v4f x1 = __builtin_nontemporal_load((const v4f*)(x + i0 + 4));

  // Split 1/(2*pi) for accurate float-only range reduction to revolutions.
  constexpr double INV2PI_D = 0.15915494309189535;
  constexpr float  INV2PI_HI = (float)INV2PI_D;
  constexpr float  INV2PI_LO = (float)(INV2PI_D - (double)INV2PI_HI);

  float sv[4], cv[4];
#pragma unroll
  for (int j = 0; j < 4; ++j) {
    // Mirror the reference: arg = float(2k) * float(-ln(1e4)/D); dt = exp(arg)
    const float arg = (float)(d0 + 2 * j) * negLogOverD;
    const float dt  = expf(arg);                 // accurate v_exp_f32 path
    const float ang = posf * dt;                 // radians, up to ~1e4

    // ang -> revolutions with extended precision: f in [-0.5, 0.5]
    const float p  = ang * INV2PI_HI;
    const float e  = fmaf(ang, INV2PI_HI, -p);   // exact product residual
    const float f0 = p - rintf(p);               // exact (Sterbenz)
    const float f  = f0 + fmaf(ang, INV2PI_LO, e);

    // v_sin_f32 / v_cos_f32 take the angle pre-scaled by 1/(2*pi).
    sv[j] = __builtin_amdgcn_sinf(f);
    cv[j] = __builtin_amdgcn_cosf(f);
  }

  v4f y0, y1;
  y0.x = fmaf(x0.x, sqrtD, sv[0]);
  y0.y = fmaf(x0.y, sqrtD, cv[0]);
  y0.z = fmaf(x0.z, sqrtD, sv[1]);
  y0.w = fmaf(x0.w, sqrtD, cv[1]);
  y1.x = fmaf(x1.x, sqrtD, sv[2]);
  y1.y = fmaf(x1.y, sqrtD, cv[2]);
  y1.z = fmaf(x1.z, sqrtD, sv[3]);
  y1.w = fmaf(x1.w, sqrtD, cv[3]);

  __builtin_nontemporal_store(y0, (v4f*)(out + i0));
  __builtin_nontemporal_store(y1, (v4f*)(out + i0 + 4));
}

extern "C" void kernel_launch(void* const* d_in, const int* in_sizes, int n_in,
                              void* d_out, int out_size, void* d_ws, size_t ws_size,
                              hipStream_t stream) {
  (void)n_in; (void)d_ws; (void)ws_size; (void)out_size;

  const float* x         = (const float*)d_in[0];  // [B, S, D] f32
  const int*   positions = (const int*)d_in[1];    // [B, S]    i32
  float*       out       = (float*)d_out;          // [B, S, D] f32

  const long long n    = (long long)in_sizes[0];   // B*S*D = 33,554,432
  const int       rows = in_sizes[1];              // B*S
  const int       d    = (int)(n / rows);          // D = 512 (power of two)

  int log2d = 0;
  while ((1 << (log2d + 1)) <= d) ++log2d;         // d == 1 << log2d
  const int dmask = d - 1;

  const float negLogOverD = (float)(-log(10000.0) / (double)d);
  const float sqrtD       = sqrtf((float)d);

  const long long perBlock = (long long)PE_THREADS * PE_VEC;   // 2048 floats
  const int blocks = (int)((n + perBlock - 1) / perBlock);     // 16384

  PositionalEncoding_gfx1250_kernel<<<blocks, PE_THREADS, 0, stream>>>(
      x, positions, out, n, log2d, dmask, negLogOverD, sqrtD);
}